// RPN_82489141887721
// MI455X (gfx1250) — compile-verified
//
#include <hip/hip_runtime.h>
#include <hip/hip_bf16.h>

// RPN anchor generation + delta application + IoU matching for MI455X (gfx1250).
// Memory-bound (~2.8 MB total traffic); CDNA5 features used:
//  - async global->LDS DMA (ASYNCcnt) to stage the 64x5 gt-box table per block
//  - TRANS32 hw transcendentals (v_exp_f32 / v_log_f32) that co-execute with VALU
//  - LDS broadcast reads in the 64-iteration match loop (conflict-free)

#define NEGV (-100000000.0f)

namespace {
constexpr int kN3 = 30000;   // p3 anchors (10000 locs * 3 AR)
constexpr int kN4 = 7500;    // p4 anchors (2500 locs * 3 AR)
constexpr int kN5 = 1875;    // p5 anchors (625 locs * 3 AR)
constexpr int kNA = kN3 + kN4 + kN5;  // 39375
constexpr int kNG = 64;      // gt boxes
constexpr float kScaleClamp = 3.3322045f;       // log(224/8)
constexpr float kRs2  = 1.41421356237f;         // sqrt(2)
constexpr float kIrs2 = 0.70710678119f;         // 1/sqrt(2)
}

__global__ __launch_bounds__(256) void rpn_match_kernel(
    const float* __restrict__ loc3,
    const float* __restrict__ loc4,
    const float* __restrict__ loc5,
    const float* __restrict__ deltas,
    const float* __restrict__ gt,     // 64 x 5
    float* __restrict__ out)
{
    __shared__ float sgt[kNG * 5];    // 320 floats = 80 x b128
    __shared__ float sarea[kNG];      // hoisted per-gt areas

    const int tid = threadIdx.x;

    // ---- CDNA5 async DMA: stage gt boxes into LDS (ASYNCcnt path) ----
    if (tid < 80) {
        // low 32 bits of the generic pointer to LDS == LDS byte offset
        unsigned ldsOff = (unsigned)(unsigned long long)(const void*)&sgt[tid * 4];
        unsigned long long ga = (unsigned long long)(const void*)(gt + tid * 4);
        asm volatile("global_load_async_to_lds_b128 %0, %1, off"
                     :: "v"(ldsOff), "v"(ga) : "memory");
    }
    asm volatile("s_wait_asynccnt 0x0" ::: "memory");
    __syncthreads();

    // ---- hoist loop-invariant gt areas (shared by all threads) ----
    if (tid < kNG) {
        const float gx0 = sgt[tid * 5 + 0];
        const float gy0 = sgt[tid * 5 + 1];
        const float gx1 = sgt[tid * 5 + 2];
        const float gy1 = sgt[tid * 5 + 3];
        sarea[tid] = (gx1 - gx0) * (gy1 - gy0);
    }
    __syncthreads();

    const int a = blockIdx.x * blockDim.x + tid;
    if (a >= kNA) return;

    // ---- reconstruct anchor analytically (branchless level select) ----
    const bool is4 = (a >= kN3) & (a < kN3 + kN4);
    const bool is5 = (a >= kN3 + kN4);
    const float* locs = is5 ? loc5 : (is4 ? loc4 : loc3);
    const int   li   = a - (is4 ? kN3 : 0) - (is5 ? (kN3 + kN4) : 0);
    const float S    = is5 ? 128.0f : (is4 ? 64.0f : 32.0f); // stride*STRIDE_SCALE

    const int loc = li / 3;
    const int ar  = li - loc * 3;
    const float x = locs[2 * loc];
    const float y = locs[2 * loc + 1];

    // AR 0.5 -> w=S*sqrt2, h=S/sqrt2 ; AR 1 -> S,S ; AR 2 -> S/sqrt2, S*sqrt2
    const float fw = (ar == 0) ? kRs2 : ((ar == 1) ? 1.0f : kIrs2);
    const float fh = (ar == 0) ? kIrs2 : ((ar == 1) ? 1.0f : kRs2);
    const float hw = 0.5f * S * fw;
    const float hh = 0.5f * S * fh;

    const float ax0 = x - hw, ay0 = y - hh, ax1 = x + hw, ay1 = y + hh;
    const float px = x, py = y;
    const float pw = ax1 - ax0, ph = ay1 - ay0;

    // ---- apply deltas -> proposals (out[0 .. 4N)) ----
    const float4 d = reinterpret_cast<const float4*>(deltas)[a];
    const float dw = fminf(d.z, kScaleClamp);
    const float dh = fminf(d.w, kScaleClamp);
    const float bx = px + pw * d.x;
    const float by = py + ph * d.y;
    const float bw = pw * __expf(dw);      // v_exp_f32 (TRANS32, co-executes)
    const float bh = ph * __expf(dh);
    float4 prop = make_float4(bx - 0.5f * bw, by - 0.5f * bh,
                              bx + 0.5f * bw, by + 0.5f * bh);
    if (d.x == NEGV) prop = make_float4(NEGV, NEGV, NEGV, NEGV);
    reinterpret_cast<float4*>(out)[a] = prop;   // base is 16B aligned

    // ---- IoU argmax over 64 gt boxes (LDS broadcast reads) ----
    const float a1 = pw * ph;
    float best = -3.0e38f;
    int  besti = 0;
#pragma unroll 8
    for (int g = 0; g < kNG; ++g) {
        const float gx0 = sgt[g * 5 + 0];
        const float gy0 = sgt[g * 5 + 1];
        const float gx1 = sgt[g * 5 + 2];
        const float gy1 = sgt[g * 5 + 3];
        const float a2  = sarea[g];
        const float ix0 = fmaxf(ax0, gx0);
        const float iy0 = fmaxf(ay0, gy0);
        const float ix1 = fminf(ax1, gx1);
        const float iy1 = fminf(ay1, gy1);
        const float inter = fmaxf(ix1 - ix0, 0.0f) * fmaxf(iy1 - iy0, 0.0f);
        const float uni   = a1 + a2 - inter;
        const float iou   = inter / uni;
        if (iou > best) { best = iou; besti = g; }   // first-max tie break
    }

    // ---- threshold masking -> matched_gt (out[4N .. 9N), row stride 5) ----
    float m0 = sgt[besti * 5 + 0];
    float m1 = sgt[besti * 5 + 1];
    float m2 = sgt[besti * 5 + 2];
    float m3 = sgt[besti * 5 + 3];
    float m4 = sgt[besti * 5 + 4];
    if (best <= 0.3f)      { m0 = m1 = m2 = m3 = m4 = -1.0f; }
    else if (best < 0.7f)  { m0 = m1 = m2 = m3 = m4 = NEGV;  }

    float* mout = out + 4 * kNA + a * 5;
    mout[0] = m0; mout[1] = m1; mout[2] = m2; mout[3] = m3; mout[4] = m4;

    // ---- gt deltas from matched box (out[9N .. 13N)) ----
    const float gbx = 0.5f * (m0 + m2);
    const float gby = 0.5f * (m1 + m3);
    const float gbw = m2 - m0;
    const float gbh = m3 - m1;
    float e0 = (gbx - px) / pw;
    float e1 = (gby - py) / ph;
    float e2 = __logf(gbw / pw);           // v_log_f32 (TRANS32)
    float e3 = __logf(gbh / ph);
    if ((m0 + m1 + m2 + m3) == -4.0f) { e0 = e1 = e2 = e3 = NEGV; }

    float* gout = out + 9 * kNA + a * 4;   // base not 16B aligned
    gout[0] = e0; gout[1] = e1; gout[2] = e2; gout[3] = e3;
}

extern "C" void kernel_launch(void* const* d_in, const int* in_sizes, int n_in,
                              void* d_out, int out_size, void* d_ws, size_t ws_size,
                              hipStream_t stream) {
    (void)in_sizes; (void)n_in; (void)out_size; (void)d_ws; (void)ws_size;
    const float* loc3   = (const float*)d_in[0];
    const float* loc4   = (const float*)d_in[1];
    const float* loc5   = (const float*)d_in[2];
    const float* deltas = (const float*)d_in[3];
    const float* gt     = (const float*)d_in[4];
    float* out = (float*)d_out;

    const int threads = 256;
    const int blocks  = (kNA + threads - 1) / threads;   // 154
    rpn_match_kernel<<<blocks, threads, 0, stream>>>(loc3, loc4, loc5,
                                                     deltas, gt, out);
}